// MY_GCN_1_6038724018515
// MI455X (gfx1250) — compile-verified
//
#include <hip/hip_runtime.h>
#include <hip/hip_bf16.h>

// ---------------------------------------------------------------------------
// GCN forward for MI455X (gfx1250, wave32).
//  - fp32 WMMA (V_WMMA_F32_16X16X4_F32): each wave computes a 16x64 output
//    strip (4 N-tiles) so the activation matrix is read fout/64 times instead
//    of fout/16 (A-traffic /4); weights are L2-resident.
//  - wave-per-edge gather/scatter with b128 gathers + hardware f32 atomics
//    (the real bottleneck: ~26 GB of mostly-L2 traffic -> ~1.1 ms @ 23.3 TB/s)
//  - two-pass BatchNorm (biased var) fused with bias + ReLU
// ---------------------------------------------------------------------------

typedef float v2f __attribute__((ext_vector_type(2)));
typedef float v8f __attribute__((ext_vector_type(8)));

__device__ __forceinline__ void atomic_add_f32(float* p, float v) {
  // lowers to global_atomic_add_f32 on gfx1250
  unsafeAtomicAdd(p, v);
}

// --------------------------- utility: zero fill ----------------------------
__global__ void zero_kernel(float* __restrict__ p, size_t n) {
  size_t i = (size_t)blockIdx.x * blockDim.x + threadIdx.x;
  size_t stride = (size_t)gridDim.x * blockDim.x;
  for (; i < n; i += stride) p[i] = 0.0f;
}

// ------------------- WMMA fp32 GEMM: C[M,Nf] = A[M,K] @ W[K,Nf] ------------
// One wave computes a 16 x (16*NT) strip; 8 waves/block -> 128 rows/block.
// Requires K % 4 == 0, Nf % (16*NT) == 0, M % 16 == 0.
template <int NT>
__global__ void wmma_gemm_kernel(const float* __restrict__ A,
                                 const float* __restrict__ W,
                                 float* __restrict__ C,
                                 int M, int K, int Nf) {
  const int wave = threadIdx.x >> 5;
  const int lane = threadIdx.x & 31;
  const int n0 = blockIdx.x * (16 * NT);
  const int m0 = (blockIdx.y * 8 + wave) * 16;
  if (m0 >= M) return;  // wave-uniform: EXEC stays all-ones for live waves

  const int lm = lane & 15;   // row/col within tile
  const int hi = lane >> 4;   // 0: K pair {k,k+1}, 1: {k+2,k+3}

  v8f c[NT];
  v8f zero = {};
#pragma unroll
  for (int t = 0; t < NT; ++t) c[t] = zero;

  const float* arow = A + (size_t)(m0 + lm) * K + hi * 2;
  for (int k = 0; k < K; k += 4) {
    v2f a = *(const v2f*)(arow + k);          // {A[r][k+2hi], A[r][k+2hi+1]}
    __builtin_prefetch(arow + k + 16, 0, 0);  // global_prefetch_b8
    const float* wp = W + (size_t)(k + hi * 2) * Nf + n0 + lm;
#pragma unroll
    for (int t = 0; t < NT; ++t) {
      v2f b;
      b.x = wp[t * 16];       // W[k+2hi  ][n0+16t+lm]
      b.y = wp[Nf + t * 16];  // W[k+2hi+1][n0+16t+lm]
      c[t] = __builtin_amdgcn_wmma_f32_16x16x4_f32(
          /*neg_a=*/false, a, /*neg_b=*/false, b,
          /*c_mod=*/(short)0, c[t], /*reuse_a=*/false, /*reuse_b=*/false);
    }
  }

  // C/D layout: VGPR v -> row m0+v (lanes 0-15) / m0+8+v (lanes 16-31)
  const int row = m0 + hi * 8;
#pragma unroll
  for (int t = 0; t < NT; ++t) {
    const int col = n0 + t * 16 + lm;
#pragma unroll
    for (int v = 0; v < 8; ++v) {
      C[(size_t)(row + v) * Nf + col] = c[t][v];
    }
  }
}

static inline void launch_gemm(const float* A, const float* Wm, float* Cout,
                               int M, int K, int Nf, hipStream_t stream) {
  const int mblocks = (M / 16 + 7) / 8;
  if (Nf % 64 == 0) {
    dim3 grid(Nf / 64, mblocks);
    wmma_gemm_kernel<4><<<grid, 256, 0, stream>>>(A, Wm, Cout, M, K, Nf);
  } else if (Nf % 32 == 0) {
    dim3 grid(Nf / 32, mblocks);
    wmma_gemm_kernel<2><<<grid, 256, 0, stream>>>(A, Wm, Cout, M, K, Nf);
  } else {
    dim3 grid(Nf / 16, mblocks);
    wmma_gemm_kernel<1><<<grid, 256, 0, stream>>>(A, Wm, Cout, M, K, Nf);
  }
}

// ------------------- layer 0 GEMM (fin=3 -> fout=16), plain ----------------
__global__ void gemm3_kernel(const float* __restrict__ X,
                             const float* __restrict__ W,
                             float* __restrict__ S, int M) {
  int i = blockIdx.x * blockDim.x + threadIdx.x;
  if (i >= M * 16) return;
  int n = i >> 4, f = i & 15;
  const float* xr = X + (size_t)n * 3;
  S[i] = xr[0] * W[f] + xr[1] * W[16 + f] + xr[2] * W[32 + f];
}

// ------------- edge scatter: agg[dst] += support[src] * w (per edge) -------
// One wave per edge; lane L covers features 4L..4L+3 (b128 gather loads),
// stride 128. Requires Nf % 4 == 0 (true for all layers).
__global__ void scatter_kernel(const float* __restrict__ S,
                               const int* __restrict__ src,
                               const int* __restrict__ dst,
                               const float* __restrict__ ew,
                               float* __restrict__ agg,
                               int nEdges, int Nf) {
  int wave = (int)(((size_t)blockIdx.x * blockDim.x + threadIdx.x) >> 5);
  int lane = threadIdx.x & 31;
  if (wave >= nEdges) return;
  int s = src[wave];
  int d = dst[wave];
  float w = ew[wave];
  const float* srow = S + (size_t)s * Nf;
  float* drow = agg + (size_t)d * Nf;
  for (int f = lane * 4; f < Nf; f += 128) {
    float4 v = *(const float4*)(srow + f);
    atomic_add_f32(&drow[f + 0], v.x * w);
    atomic_add_f32(&drow[f + 1], v.y * w);
    atomic_add_f32(&drow[f + 2], v.z * w);
    atomic_add_f32(&drow[f + 3], v.w * w);
  }
}

// ---------------- per-feature sum / sumsq of (X + bias) --------------------
// sums[0:Nf] = sum, sums[Nf:2*Nf] = sum of squares
__global__ void stats_kernel(const float* __restrict__ X,
                             const float* __restrict__ bias,
                             float* __restrict__ sums,
                             int M, int Nf, int rowsPerBlock) {
  int r0 = blockIdx.x * rowsPerBlock;
  int r1 = min(r0 + rowsPerBlock, M);
  for (int f = threadIdx.x; f < Nf; f += blockDim.x) {
    float b = bias ? bias[f] : 0.0f;
    float s = 0.0f, q = 0.0f;
    for (int r = r0; r < r1; ++r) {
      float x = X[(size_t)r * Nf + f] + b;
      s += x;
      q += x * x;
    }
    atomic_add_f32(&sums[f], s);
    atomic_add_f32(&sums[Nf + f], q);
  }
}

// --------- fold mean/var/gamma/beta/bias into per-feature scale+shift ------
__global__ void finalize_stats_kernel(const float* __restrict__ sums,
                                      const float* __restrict__ g,
                                      const float* __restrict__ be,
                                      const float* __restrict__ bias,
                                      float* __restrict__ scale,
                                      float* __restrict__ shift,
                                      int Nf, float invN) {
  int f = blockIdx.x * blockDim.x + threadIdx.x;
  if (f >= Nf) return;
  float mean = sums[f] * invN;
  float var = sums[Nf + f] * invN - mean * mean;
  float sc = g[f] * rsqrtf(var + 1e-5f);
  float b = bias ? bias[f] : 0.0f;
  scale[f] = sc;
  shift[f] = be[f] + (b - mean) * sc;  // y = x*sc + shift (x stored w/o bias)
}

// ------------------------ y = relu(x*scale + shift) ------------------------
__global__ void bn_relu_kernel(const float* __restrict__ X,
                               const float* __restrict__ scale,
                               const float* __restrict__ shift,
                               float* __restrict__ Y,
                               size_t total, int Nf) {
  size_t i = (size_t)blockIdx.x * blockDim.x + threadIdx.x;
  if (i >= total) return;
  int f = (int)(i & (size_t)(Nf - 1));  // Nf is a power of two
  float y = X[i] * scale[f] + shift[f];
  Y[i] = fmaxf(y, 0.0f);
}

// ------------------------- final FC 64 -> 5 + bias -------------------------
__global__ void fc_final_kernel(const float* __restrict__ H,
                                const float* __restrict__ W,
                                const float* __restrict__ b,
                                float* __restrict__ out, int M) {
  int i = blockIdx.x * blockDim.x + threadIdx.x;
  if (i >= M * 5) return;
  int n = i / 5, j = i - n * 5;
  const float* h = H + (size_t)n * 64;
  float acc = b[j];
#pragma unroll
  for (int k = 0; k < 64; ++k) acc += h[k] * W[k * 5 + j];
  out[i] = acc;
}

// ---------------------------------------------------------------------------
extern "C" void kernel_launch(void* const* d_in, const int* in_sizes, int n_in,
                              void* d_out, int out_size, void* d_ws, size_t ws_size,
                              hipStream_t stream) {
  const float* x = (const float*)d_in[0];
  const int* esrc = (const int*)d_in[1];
  const int* edst = (const int*)d_in[2];
  const float* ew = (const float*)d_in[3];

  // params flattened in dict-insertion order
  const float *W[6], *b[6], *g[6], *be[6];
  for (int i = 0; i < 6; ++i) {
    W[i] = (const float*)d_in[4 + 4 * i + 0];
    b[i] = (const float*)d_in[4 + 4 * i + 1];
    g[i] = (const float*)d_in[4 + 4 * i + 2];
    be[i] = (const float*)d_in[4 + 4 * i + 3];
  }
  const float *fW[4], *fb[4], *fg[3], *fbe[3];
  int idx = 28;
  for (int j = 0; j < 3; ++j) {
    fW[j] = (const float*)d_in[idx++];
    fb[j] = (const float*)d_in[idx++];
    fg[j] = (const float*)d_in[idx++];
    fbe[j] = (const float*)d_in[idx++];
  }
  fW[3] = (const float*)d_in[idx++];
  fb[3] = (const float*)d_in[idx++];

  const int M = in_sizes[0] / 3;   // 100000 (divisible by 16)
  const int E = in_sizes[1];       // 3200000

  // workspace carve-out
  float* ws = (float*)d_ws;
  const size_t NM = (size_t)M * 512;
  float* Hbuf = ws;            // current activations
  float* Sbuf = ws + NM;       // GEMM output ("support")
  float* Abuf = ws + 2 * NM;   // aggregation target
  float* sums = ws + 3 * NM;   // 2*512 stats
  float* scale = sums + 1024;  // 512
  float* shift = scale + 512;  // 512

  const int dims[7] = {3, 16, 32, 64, 128, 256, 512};

  for (int i = 0; i < 6; ++i) {
    const int fout = dims[i + 1];
    // 1) support = h @ W
    if (i == 0) {
      int tot = M * 16;
      gemm3_kernel<<<(tot + 255) / 256, 256, 0, stream>>>(x, W[0], Sbuf, M);
    } else {
      launch_gemm(Hbuf, W[i], Sbuf, M, dims[i], fout, stream);
    }
    // 2) zero aggregation buffer + stats
    zero_kernel<<<2048, 256, 0, stream>>>(Abuf, (size_t)M * fout);
    zero_kernel<<<4, 256, 0, stream>>>(sums, 1024);
    // 3) scatter: agg[dst] += support[src] * w
    int sblocks = (E + 7) / 8;  // 8 waves (edges) per 256-thread block
    scatter_kernel<<<sblocks, 256, 0, stream>>>(Sbuf, esrc, edst, ew, Abuf, E, fout);
    // 4) BN stats over (agg + bias)
    const int rpb = 500;
    stats_kernel<<<(M + rpb - 1) / rpb, 256, 0, stream>>>(Abuf, b[i], sums, M, fout, rpb);
    finalize_stats_kernel<<<(fout + 63) / 64, 64, 0, stream>>>(
        sums, g[i], be[i], b[i], scale, shift, fout, 1.0f / (float)M);
    // 5) h = relu(bn(agg + bias))
    size_t total = (size_t)M * fout;
    bn_relu_kernel<<<(int)((total + 255) / 256), 256, 0, stream>>>(
        Abuf, scale, shift, Hbuf, total, fout);
  }

  const int fdims[5] = {512, 256, 128, 64, 5};
  for (int j = 0; j < 3; ++j) {
    const int fout = fdims[j + 1];
    launch_gemm(Hbuf, fW[j], Sbuf, M, fdims[j], fout, stream);
    zero_kernel<<<4, 256, 0, stream>>>(sums, 1024);
    const int rpb = 500;
    stats_kernel<<<(M + rpb - 1) / rpb, 256, 0, stream>>>(Sbuf, fb[j], sums, M, fout, rpb);
    finalize_stats_kernel<<<(fout + 63) / 64, 64, 0, stream>>>(
        sums, fg[j], fbe[j], fb[j], scale, shift, fout, 1.0f / (float)M);
    size_t total = (size_t)M * fout;
    bn_relu_kernel<<<(int)((total + 255) / 256), 256, 0, stream>>>(
        Sbuf, scale, shift, Hbuf, total, fout);
  }

  // final linear 64 -> 5
  fc_final_kernel<<<(M * 5 + 255) / 256, 256, 0, stream>>>(
      Hbuf, fW[3], fb[3], (float*)d_out, M);
}